// BaselineModel_38981123178598
// MI455X (gfx1250) — compile-verified
//
#include <hip/hip_runtime.h>
#include <math.h>

typedef __attribute__((ext_vector_type(2))) float v2f;
typedef __attribute__((ext_vector_type(8))) float v8f;

#define HID 128
#define NGRAPH 256
#define NCLS 10
#define NLAYERS 3
#define LN_EPS 1e-5f

// ---------------- small utility kernels ----------------

__global__ void k_fill(float* __restrict__ p, float v, long n) {
  long i = (long)blockIdx.x * blockDim.x + threadIdx.x;
  if (i < n) p[i] = v;
}

__global__ void k_edge_deg(const int* __restrict__ dst, float* __restrict__ deg, int E) {
  int e = blockIdx.x * blockDim.x + threadIdx.x;
  if (e < E) atomicAdd(&deg[dst[e]], 1.0f);
}

__global__ void k_deg2dinv(float* __restrict__ d, int N) {
  int i = blockIdx.x * blockDim.x + threadIdx.x;
  if (i < N) d[i] = rsqrtf(d[i]);   // deg >= 1 always (self-loop)
}

// ---------------- WMMA fp32 GEMM: C[M x 128] = act(A[M x 128] @ W[128 x 128] + bias) ----------------
// REQUIRES M % 16 == 0 (true for all call sites: 100000 and 256).
// One block = 16 output rows; 8 waves, each wave owns a 16x16 output tile (16 cols).
// Uses V_WMMA_F32_16X16X4_F32, K-loop of 32 steps (unrolled 8x).
template <bool RELU, bool HAS_BIAS>
__launch_bounds__(256)
__global__ void k_gemm128_wmma(const float* __restrict__ A, const float* __restrict__ W,
                               const float* __restrict__ bias, float* __restrict__ C) {
  __shared__ float lA[16 * 132];          // 16 rows, stride 132 (pad avoids bank conflicts)
  const int t = threadIdx.x;
  const long row0 = (long)blockIdx.x * 16;

  // Stage A tile (16x128) into LDS; 512 float4 loads across 256 threads.
  #pragma unroll
  for (int i = t; i < 512; i += 256) {
    int r = i >> 5;
    int c4 = (i & 31) * 4;
    float4 v = *(const float4*)(A + (row0 + r) * HID + c4);
    *(float4*)(&lA[r * 132 + c4]) = v;
  }
  __syncthreads();

  const int lane = t & 31;
  const int wave = t >> 5;                // 0..7 -> 16-column tile
  const int m    = lane & 15;
  const int hi   = lane >> 4;             // lane half
  const int n    = wave * 16 + (lane & 15);

  v8f acc = {};
  #pragma unroll 8
  for (int k0 = 0; k0 < HID; k0 += 4) {
    // A 16x4 fragment (ISA layout: lane half selects K pair)
    v2f a = *(const v2f*)(&lA[m * 132 + k0 + hi * 2]);
    // B 4x16 fragment: VGPR0 = rows k0,k0+1 per lane-half; VGPR1 = rows k0+2,k0+3
    v2f b;
    b.x = W[(k0 + hi) * HID + n];
    b.y = W[(k0 + hi + 2) * HID + n];
    acc = __builtin_amdgcn_wmma_f32_16x16x4_f32(false, a, false, b,
                                                (short)0, acc, false, false);
  }

  const float bv = HAS_BIAS ? bias[n] : 0.0f;
  #pragma unroll
  for (int v = 0; v < 8; ++v) {
    long r = row0 + v + hi * 8;           // C/D layout: M = vgpr + 8*(lane>=16)
    float val = acc[v] + bv;
    if (RELU) val = fmaxf(val, 0.0f);
    C[r * HID + n] = val;
  }
}

// ---------------- GCN aggregation ----------------

// h[i, c] = hw[i, c] * dinv[i]^2 + bias[c]   (self-loop term + bias; fully overwrites h)
__global__ void k_self_bias(const float* __restrict__ hw, const float* __restrict__ dinv,
                            const float* __restrict__ bias, float* __restrict__ h, long NH) {
  long i = (long)blockIdx.x * blockDim.x + threadIdx.x;
  if (i < NH) {
    int r = (int)(i >> 7);
    int c = (int)(i & 127);
    float di = dinv[r];
    h[i] = hw[i] * di * di + bias[c];
  }
}

// One wave per edge: float4 gather of hw[src], scaled atomic scatter into h[dst].
// Working set (h + hw ~= 102 MB) fits in the 192 MB L2, so keep default RT caching.
__global__ void k_scatter(const int* __restrict__ src, const int* __restrict__ dst,
                          const float* __restrict__ dinv, const float* __restrict__ hw,
                          float* __restrict__ h, int E) {
  long gid = (long)blockIdx.x * blockDim.x + threadIdx.x;
  int e = (int)(gid >> 5);
  int lane = (int)(gid & 31);
  if (e >= E) return;
  int s = src[e], d = dst[e];
  float coeff = dinv[s] * dinv[d];
  const float4 v = *(const float4*)(hw + (long)s * HID + lane * 4);
  float* o = h + (long)d * HID + lane * 4;
  atomicAdd(o + 0, v.x * coeff);
  atomicAdd(o + 1, v.y * coeff);
  atomicAdd(o + 2, v.z * coeff);
  atomicAdd(o + 3, v.w * coeff);
}

// ---------------- full-tensor LayerNorm ----------------

__global__ void k_zero_stats(double* __restrict__ stats) {
  stats[0] = 0.0; stats[1] = 0.0;
}

__global__ void k_relu_reduce(float* __restrict__ h, double* __restrict__ stats, long NH) {
  __shared__ double ssum[256];
  __shared__ double ssq[256];
  int t = threadIdx.x;
  double ps = 0.0, pq = 0.0;
  for (long i = (long)blockIdx.x * blockDim.x + t; i < NH; i += (long)gridDim.x * blockDim.x) {
    float v = fmaxf(h[i], 0.0f);
    h[i] = v;
    ps += (double)v;
    pq += (double)v * (double)v;
  }
  ssum[t] = ps; ssq[t] = pq;
  __syncthreads();
  for (int s = 128; s > 0; s >>= 1) {
    if (t < s) { ssum[t] += ssum[t + s]; ssq[t] += ssq[t + s]; }
    __syncthreads();
  }
  if (t == 0) {
    atomicAdd(&stats[0], ssum[0]);
    atomicAdd(&stats[1], ssq[0]);
  }
}

__global__ void k_finalize_stats(const double* __restrict__ stats, float* __restrict__ statsf, long NH) {
  double inv = 1.0 / (double)NH;
  double mu = stats[0] * inv;
  double var = stats[1] * inv - mu * mu;
  statsf[0] = (float)mu;
  statsf[1] = (float)(1.0 / sqrt(var + (double)LN_EPS));
}

__global__ void k_normalize(float* __restrict__ h, const float* __restrict__ statsf, long NH) {
  long i = (long)blockIdx.x * blockDim.x + threadIdx.x;
  if (i < NH) h[i] = (h[i] - statsf[0]) * statsf[1];
}

// ---------------- pooling + MLP head ----------------

__global__ void k_pool(const float* __restrict__ h, const int* __restrict__ batch,
                       float* __restrict__ pooled, float* __restrict__ cnt, int N) {
  long gid = (long)blockIdx.x * blockDim.x + threadIdx.x;
  int node = (int)(gid >> 5);
  int lane = (int)(gid & 31);
  if (node >= N) return;
  int g = batch[node];
  const float4 v = *(const float4*)(h + (long)node * HID + lane * 4);
  float* o = pooled + (long)g * HID + lane * 4;
  atomicAdd(o + 0, v.x);
  atomicAdd(o + 1, v.y);
  atomicAdd(o + 2, v.z);
  atomicAdd(o + 3, v.w);
  if (lane == 0) atomicAdd(&cnt[g], 1.0f);
}

__global__ void k_pool_div(float* __restrict__ pooled, const float* __restrict__ cnt) {
  int i = blockIdx.x * blockDim.x + threadIdx.x;
  if (i < NGRAPH * HID) pooled[i] /= fmaxf(cnt[i >> 7], 1.0f);
}

// 1 wave per graph: scores = g1 @ W2 + b2, then log_softmax.
__global__ void k_head(const float* __restrict__ g1, const float* __restrict__ W2,
                       const float* __restrict__ b2, float* __restrict__ out) {
  int r = blockIdx.x;
  int lane = threadIdx.x;                  // 32 lanes
  const float4 v = *(const float4*)(g1 + (long)r * HID + lane * 4);
  float sc[NCLS];
  #pragma unroll
  for (int c = 0; c < NCLS; ++c) {
    float p = v.x * W2[(lane * 4 + 0) * NCLS + c]
            + v.y * W2[(lane * 4 + 1) * NCLS + c]
            + v.z * W2[(lane * 4 + 2) * NCLS + c]
            + v.w * W2[(lane * 4 + 3) * NCLS + c];
    #pragma unroll
    for (int off = 16; off >= 1; off >>= 1) p += __shfl_xor(p, off, 32);
    sc[c] = p + b2[c];                     // all lanes hold the full sum
  }
  float mx = sc[0];
  #pragma unroll
  for (int c = 1; c < NCLS; ++c) mx = fmaxf(mx, sc[c]);
  float se = 0.0f;
  #pragma unroll
  for (int c = 0; c < NCLS; ++c) se += __expf(sc[c] - mx);
  float lse = mx + __logf(se);
  if (lane == 0) {
    #pragma unroll
    for (int c = 0; c < NCLS; ++c) out[(long)r * NCLS + c] = sc[c] - lse;
  }
}

// ---------------- launch ----------------

extern "C" void kernel_launch(void* const* d_in, const int* in_sizes, int n_in,
                              void* d_out, int out_size, void* d_ws, size_t ws_size,
                              hipStream_t stream) {
  const float* x      = (const float*)d_in[0];
  const int*   ei     = (const int*)  d_in[1];
  const int*   batch  = (const int*)  d_in[2];
  const float* lin1_W = (const float*)d_in[3];
  const float* lin1_b = (const float*)d_in[4];
  const float* conv_W = (const float*)d_in[5];
  const float* conv_b = (const float*)d_in[6];
  const float* mlp_W1 = (const float*)d_in[7];
  const float* mlp_b1 = (const float*)d_in[8];
  const float* mlp_W2 = (const float*)d_in[9];
  const float* mlp_b2 = (const float*)d_in[10];
  float* out = (float*)d_out;

  const int N = in_sizes[0] / HID;   // 100000 (multiple of 16)
  const int E = in_sizes[1] / 2;
  const int* src = ei;
  const int* dst = ei + E;
  const long NH = (long)N * HID;

  // workspace layout
  char* w = (char*)d_ws;
  double* stats  = (double*)w;                         // 2 doubles
  float*  statsf = (float*)(w + 16);                   // 2 floats
  float*  h      = (float*)(w + 32);                   // N*HID
  float*  hw     = h + NH;                             // N*HID
  float*  dinv   = hw + NH;                            // N
  float*  pooled = dinv + N;                           // NGRAPH*HID
  float*  cnt    = pooled + (long)NGRAPH * HID;        // NGRAPH (contiguous after pooled)
  float*  g1     = cnt + NGRAPH;                       // NGRAPH*HID

  const int T = 256;
  dim3 blk(T);

  // 1) degrees with self-loops -> dinv
  k_fill<<<(N + T - 1) / T, blk, 0, stream>>>(dinv, 1.0f, N);
  k_edge_deg<<<(E + T - 1) / T, blk, 0, stream>>>(dst, dinv, E);
  k_deg2dinv<<<(N + T - 1) / T, blk, 0, stream>>>(dinv, N);

  // 2) h = relu(x @ lin1_W + b)
  k_gemm128_wmma<true, true><<<N / 16, blk, 0, stream>>>(x, lin1_W, lin1_b, h);

  // 3) GCN layers
  for (int l = 0; l < NLAYERS; ++l) {
    k_gemm128_wmma<false, false><<<N / 16, blk, 0, stream>>>(h, conv_W + (long)l * HID * HID,
                                                             nullptr, hw);
    k_self_bias<<<(unsigned)((NH + T - 1) / T), blk, 0, stream>>>(hw, dinv, conv_b + l * HID, h, NH);
    k_scatter<<<(unsigned)(((long)E * 32 + T - 1) / T), blk, 0, stream>>>(src, dst, dinv, hw, h, E);
    k_zero_stats<<<1, 1, 0, stream>>>(stats);
    k_relu_reduce<<<1024, blk, 0, stream>>>(h, stats, NH);
    k_finalize_stats<<<1, 1, 0, stream>>>(stats, statsf, NH);
    k_normalize<<<(unsigned)((NH + T - 1) / T), blk, 0, stream>>>(h, statsf, NH);
  }

  // 4) global mean pooling (pooled and cnt are contiguous -> one zero fill)
  k_fill<<<(NGRAPH * (HID + 1) + T - 1) / T, blk, 0, stream>>>(pooled, 0.0f, (long)NGRAPH * (HID + 1));
  k_pool<<<(unsigned)(((long)N * 32 + T - 1) / T), blk, 0, stream>>>(h, batch, pooled, cnt, N);
  k_pool_div<<<(NGRAPH * HID + T - 1) / T, blk, 0, stream>>>(pooled, cnt);

  // 5) MLP head
  k_gemm128_wmma<true, true><<<NGRAPH / 16, blk, 0, stream>>>(pooled, mlp_W1, mlp_b1, g1);
  k_head<<<NGRAPH, 32, 0, stream>>>(g1, mlp_W2, mlp_b2, out);
}